// Grail_37950331028075
// MI455X (gfx1250) — compile-verified
//
#include <hip/hip_runtime.h>
#include <hip/hip_bf16.h>

// ---------------------------------------------------------------------------
// GraIL-style RGCN forward for MI455X (gfx1250, wave32, WMMA).
// Strategy: factor the edge-attention MLP into per-node / per-relation
// projections computed by one fused f32 WMMA GEMM [N,32]@[32,224] per layer,
// then a gather + LDS-segment-sum edge kernel (all gather targets L2-resident).
// ---------------------------------------------------------------------------

#define NNODES   100000
#define NGRAPHS  100
#define NPER     1000
#define EPG      16000      // edges per graph
#define D        32
#define NREL     200
#define NBASES   4
#define NLAYERS  3
#define PCOLS    224        // 128 xproj | 32 srcpart | 32 dstpart | 32 selfloop
#define NTILES   (PCOLS/16) // 14
#define MTILES   (NNODES/16)// 6250
#define RANGES   4
#define RNODES   (NPER/RANGES) // 250

typedef float v2f __attribute__((ext_vector_type(2)));
typedef float v8f __attribute__((ext_vector_type(8)));

// ---- pack fused weight [32,224] and relation projections [200,32]x2 --------
__global__ __launch_bounds__(256) void grail_prep(
    const float* __restrict__ basis_l,   // [4,32,32]
    const float* __restrict__ A_w_l,     // [128,32]
    const float* __restrict__ slw_l,     // [32,32]
    const float* __restrict__ attn_tab,  // [200,32]
    float* __restrict__ Wcat,            // [32,224]
    float* __restrict__ relA,            // [200,32]
    float* __restrict__ relB)            // [200,32]
{
    int i = blockIdx.x * blockDim.x + threadIdx.x;
    if (i < 32 * PCOLS) {
        int d = i / PCOLS, j = i % PCOLS;
        float v;
        if (j < 128)      { int b = j >> 5, o = j & 31; v = basis_l[b*1024 + d*32 + o]; }
        else if (j < 160) { v = A_w_l[d*32 + (j-128)]; }          // A_w rows 0..31 (src)
        else if (j < 192) { v = A_w_l[(32 + d)*32 + (j-160)]; }   // A_w rows 32..63 (dst)
        else              { v = slw_l[d*32 + (j-192)]; }          // self-loop
        Wcat[i] = v;
    } else {
        int k = i - 32 * PCOLS;
        if (k < NREL * 64) {
            int t = k >> 6, c = k & 63, o = c & 31;
            const float* arow = A_w_l + (c < 32 ? 64 : 96) * 32;  // rows 64.. / 96..
            float s = 0.f;
            #pragma unroll
            for (int j = 0; j < 32; ++j) s += attn_tab[t*32 + j] * arow[j*32 + o];
            (c < 32 ? relA : relB)[t*32 + o] = s;
        }
    }
}

// ---- fused node GEMM: P[N,224] = h[N,32] @ Wcat[32,224] via WMMA f32 -------
__global__ __launch_bounds__(256) void grail_gemm_wmma(
    const float* __restrict__ h, int hstride,
    const float* __restrict__ W,   // [32,224]
    float* __restrict__ P)         // [N,224]
{
    int wave  = (blockIdx.x << 3) + (threadIdx.x >> 5);
    int mtile = wave / NTILES;
    int ntile = wave % NTILES;
    if (mtile >= MTILES) return;          // uniform per wave: EXEC stays all-ones
    int lane = threadIdx.x & 31;
    int m    = lane & 15;
    int half = lane >> 4;                 // 0 | 1
    const float* __restrict__ arow = h + (size_t)(mtile*16 + m) * hstride;
    int col = ntile*16 + m;               // B / output column (lane&15)
    v8f c = {0.f,0.f,0.f,0.f,0.f,0.f,0.f,0.f};
    #pragma unroll
    for (int ks = 0; ks < 8; ++ks) {
        int k = ks*4 + half*2;            // K = k + v per ISA 16x4 f32 A layout
        v2f a; a.x = arow[k];             a.y = arow[k+1];
        v2f b; b.x = W[k*PCOLS + col];    b.y = W[(k+1)*PCOLS + col];
        c = __builtin_amdgcn_wmma_f32_16x16x4_f32(false, a, false, b,
                                                  (short)0, c, false, false);
    }
    size_t outbase = (size_t)(mtile*16) * PCOLS + col;
    #pragma unroll
    for (int v = 0; v < 8; ++v)           // C/D: row = v + 8*half
        P[outbase + (size_t)(v + 8*half) * PCOLS] = c[v];
}

// ---- edge kernel: attention + weighted message, LDS segment-sum ------------
__global__ __launch_bounds__(256) void grail_edges(
    const int* __restrict__ src, const int* __restrict__ dst,
    const int* __restrict__ etype, const int* __restrict__ elabel,
    const float* __restrict__ P,          // [N,224]
    const float* __restrict__ relA, const float* __restrict__ relB,
    const float* __restrict__ A_b_l,      // [32]
    const float* __restrict__ B_w_l,      // [32]
    const float* __restrict__ B_b_l,      // [1]
    const float* __restrict__ w_comp_l,   // [200,4]
    float* __restrict__ nei)              // [N,32]
{
    __shared__ float acc[RNODES * 32];    // 32 KB accumulator for 250 nodes
    int g        = blockIdx.x / RANGES;
    int rng      = blockIdx.x % RANGES;
    int nodeBase = g * NPER + rng * RNODES;
    for (int i = threadIdx.x; i < RNODES * 32; i += blockDim.x) acc[i] = 0.f;
    __syncthreads();

    int lane = threadIdx.x & 31;
    int wave = threadIdx.x >> 5;          // 8 waves/block
    long eBase = (long)g * EPG;
    float bw = B_w_l[lane];
    float bb = B_b_l[0];
    float ab = A_b_l[lane];

    for (int e = wave; e < EPG; e += 8) {
        int d  = dst[eBase + e];
        int dl = d - nodeBase;
        if ((unsigned)dl >= (unsigned)RNODES) continue;   // uniform per wave
        int s  = src[eBase + e];
        int t  = etype[eBase + e];
        int lb = elabel[eBase + e];
        const float* __restrict__ Ps = P + (size_t)s * PCOLS;
        const float* __restrict__ Pd = P + (size_t)d * PCOLS;
        // z = relu(srcpart + dstpart + relA[etype] + relB[elabel] + A_b)
        float z = Ps[128 + lane] + Pd[160 + lane]
                + relA[t*32 + lane] + relB[lb*32 + lane] + ab;
        z = fmaxf(z, 0.f);
        float v = z * bw;
        #pragma unroll
        for (int off = 16; off; off >>= 1) v += __shfl_xor(v, off, 32);
        float a = 1.f / (1.f + __expf(-(v + bb)));        // sigmoid
        // msg = sum_b w_comp[etype,b] * xproj[src,b,:]
        float c0 = w_comp_l[t*4+0], c1 = w_comp_l[t*4+1];
        float c2 = w_comp_l[t*4+2], c3 = w_comp_l[t*4+3];
        float m = c0*Ps[lane] + c1*Ps[32+lane] + c2*Ps[64+lane] + c3*Ps[96+lane];
        atomicAdd(&acc[dl*32 + lane], a * m);             // ds_add_f32
    }
    __syncthreads();
    for (int i = threadIdx.x; i < RNODES * 32; i += blockDim.x)
        nei[(size_t)nodeBase * 32 + i] = acc[i];          // ranges tile N: plain stores
}

// ---- h = relu(nei + selfloop); write into repr[:, l, :] --------------------
__global__ __launch_bounds__(256) void grail_update(
    const float* __restrict__ P, const float* __restrict__ nei,
    float* __restrict__ repr, int l)
{
    int idx = blockIdx.x * blockDim.x + threadIdx.x;      // N*32 exactly
    int n = idx >> 5, d = idx & 31;
    float v = nei[idx] + P[(size_t)n * PCOLS + 192 + d];
    repr[(size_t)n * (NLAYERS*32) + l*32 + d] = fmaxf(v, 0.f);
}

// ---- pooling (mean over graph) + head/tail gather + final FC ---------------
__global__ __launch_bounds__(128) void grail_final(
    const float* __restrict__ repr,       // [N, 3, 32]
    const float* __restrict__ rel_tab,    // [200,32]
    const float* __restrict__ fc_w,       // [320]
    const float* __restrict__ fc_b,       // [1]
    const int* __restrict__ head_ids, const int* __restrict__ tail_ids,
    const int* __restrict__ rel_labels,
    float* __restrict__ out)              // [100]
{
    __shared__ float red[128];
    int g = blockIdx.x, t = threadIdx.x;
    float part = 0.f;
    if (t < 96) {
        float s = 0.f;
        size_t base = (size_t)g * NPER * 96 + t;
        for (int n = 0; n < NPER; ++n) s += repr[base + (size_t)n * 96];
        part += (s * (1.f / NPER)) * fc_w[t];
        int hid = head_ids[g], tid = tail_ids[g];
        part += repr[(size_t)hid * 96 + t] * fc_w[96  + t];
        part += repr[(size_t)tid * 96 + t] * fc_w[192 + t];
        if (t < 32) part += rel_tab[rel_labels[g]*32 + t] * fc_w[288 + t];
    }
    red[t] = part;
    __syncthreads();
    for (int s = 64; s; s >>= 1) { if (t < s) red[t] += red[t + s]; __syncthreads(); }
    if (t == 0) out[g] = red[0] + fc_b[0];
}

extern "C" void kernel_launch(void* const* d_in, const int* in_sizes, int n_in,
                              void* d_out, int out_size, void* d_ws, size_t ws_size,
                              hipStream_t stream) {
    const float* feat     = (const float*)d_in[0];
    const float* basis    = (const float*)d_in[1];   // [3,4,32,32]
    const float* w_comp   = (const float*)d_in[2];   // [3,200,4]
    const float* slw      = (const float*)d_in[3];   // [3,32,32]
    const float* A_w      = (const float*)d_in[4];   // [3,128,32]
    const float* A_b      = (const float*)d_in[5];   // [3,32]
    const float* B_w      = (const float*)d_in[6];   // [3,32,1]
    const float* B_b      = (const float*)d_in[7];   // [3,1]
    const float* attn_tab = (const float*)d_in[8];
    const float* rel_tab  = (const float*)d_in[9];
    const float* fc_w     = (const float*)d_in[10];
    const float* fc_b     = (const float*)d_in[11];
    const int*   src      = (const int*)d_in[12];
    const int*   dst      = (const int*)d_in[13];
    const int*   etype    = (const int*)d_in[14];
    const int*   elabel   = (const int*)d_in[15];
    const int*   head_ids = (const int*)d_in[17];
    const int*   tail_ids = (const int*)d_in[18];
    const int*   rel_lab  = (const int*)d_in[19];
    float* out = (float*)d_out;

    float* ws   = (float*)d_ws;           // ~141 MB of f32 (L2-resident)
    float* Wcat = ws;                     // 7168
    float* relA = ws + 7168;              // 6400
    float* relB = ws + 13568;             // 6400
    float* P    = ws + 19968;             // N*224 = 22,400,000
    float* nei  = ws + 22419968;          // N*32  =  3,200,000
    float* repr = ws + 25619968;          // N*96  =  9,600,000

    const int gemmBlocks = (MTILES * NTILES + 7) / 8;   // 8 waves per 256-thr block

    for (int l = 0; l < NLAYERS; ++l) {
        grail_prep<<<78, 256, 0, stream>>>(basis + l*4096, A_w + l*4096,
                                           slw + l*1024, attn_tab, Wcat, relA, relB);
        const float* h = (l == 0) ? feat : (repr + (l - 1) * 32);
        int hstride    = (l == 0) ? 32 : 96;
        grail_gemm_wmma<<<gemmBlocks, 256, 0, stream>>>(h, hstride, Wcat, P);
        grail_edges<<<NGRAPHS * RANGES, 256, 0, stream>>>(src, dst, etype, elabel,
                                                          P, relA, relB,
                                                          A_b + l*32, B_w + l*32, B_b + l,
                                                          w_comp + l*800, nei);
        grail_update<<<(NNODES * 32) / 256, 256, 0, stream>>>(P, nei, repr, l);
    }
    grail_final<<<NGRAPHS, 128, 0, stream>>>(repr, rel_tab, fc_w, fc_b,
                                             head_ids, tail_ids, rel_lab, out);
}